// GNNNet_83906481094709
// MI455X (gfx1250) — compile-verified
//
#include <hip/hip_runtime.h>
#include <math.h>

#define N_NODES  100000
#define N_EDGES  1600000
#define N_GRAPHS 512
#define F_IN     32

typedef __attribute__((ext_vector_type(2))) float v2f;
typedef __attribute__((ext_vector_type(8))) float v8f;

__device__ __forceinline__ float elu1(float v) {
    return v > 0.0f ? v : (expf(v) - 1.0f);
}

// ---------------- zero fill ----------------
__global__ void zero_f32(float* __restrict__ p, int n) {
    int i = blockIdx.x * blockDim.x + threadIdx.x;
    if (i < n) p[i] = 0.0f;
}

// ------------- edge scatter-add: agg[dst] += feat[src] -------------
// one thread per (edge, feature); features contiguous across lanes ->
// coalesced gather from feat[src] and coalesced atomics into agg[dst].
// agg buffer (<=25.6MB) lives in L2, so atomics resolve at L2 speed.
__global__ void scatter_add(const float* __restrict__ feat, float* __restrict__ agg,
                            const int* __restrict__ src, const int* __restrict__ dst,
                            int n_edges, int d_log2) {
    int idx = blockIdx.x * blockDim.x + threadIdx.x;
    int total = n_edges << d_log2;
    if (idx >= total) return;
    int e = idx >> d_log2;
    int f = idx & ((1 << d_log2) - 1);
    int s = src[e];
    int t = dst[e];
    atomicAdd(&agg[(t << d_log2) + f], feat[(s << d_log2) + f]);
}

// ------------- fused GraphConv compute: out = elu(agg@Wrel + b + hin@Wroot) -------------
// One wave32 per 16x16 output tile; K-loop over both operand pairs accumulates
// into a single f32 16x16 accumulator via v_wmma_f32_16x16x4_f32.
template <int D_IN, int D_OUT>
__global__ __launch_bounds__(256) void conv_wmma(
        const float* __restrict__ agg, const float* __restrict__ hin,
        const float* __restrict__ w_rel, const float* __restrict__ w_root,
        const float* __restrict__ bias, float* __restrict__ out, int n_nodes) {
    const int lane   = threadIdx.x & 31;
    const int wavein = threadIdx.x >> 5;
    const int gw     = blockIdx.x * (blockDim.x >> 5) + wavein;

    const int tiles_n = D_OUT / 16;
    const int tiles_m = n_nodes / 16;          // N_NODES % 16 == 0
    const int tile_m  = gw / tiles_n;
    const int tile_n  = gw % tiles_n;
    if (tile_m >= tiles_m) return;             // wave-uniform: EXEC stays all-1s below

    const int m    = lane & 15;
    const int kb   = (lane >> 4) << 1;         // 0 for lanes 0-15, 2 for lanes 16-31
    const int node = tile_m * 16 + m;          // A-fragment row
    const int col  = tile_n * 16 + m;          // B/C-fragment column

    v8f acc = {};

    // agg @ w_rel
    #pragma unroll
    for (int k = 0; k < D_IN; k += 4) {
        v2f a, b;
        a.x = agg[node * D_IN + k + kb];
        a.y = agg[node * D_IN + k + kb + 1];
        b.x = w_rel[(k + kb)     * D_OUT + col];
        b.y = w_rel[(k + kb + 1) * D_OUT + col];
        acc = __builtin_amdgcn_wmma_f32_16x16x4_f32(false, a, false, b,
                                                    (short)0, acc, false, false);
    }
    // + hin @ w_root  (accumulate into the same tile)
    #pragma unroll
    for (int k = 0; k < D_IN; k += 4) {
        v2f a, b;
        a.x = hin[node * D_IN + k + kb];
        a.y = hin[node * D_IN + k + kb + 1];
        b.x = w_root[(k + kb)     * D_OUT + col];
        b.y = w_root[(k + kb + 1) * D_OUT + col];
        acc = __builtin_amdgcn_wmma_f32_16x16x4_f32(false, a, false, b,
                                                    (short)0, acc, false, false);
    }

    // epilogue: bias + ELU, C/D layout: VGPR i holds row i + 8*(lane>>4)
    const float bj    = bias[col];
    const int   rbase = (lane >> 4) * 8;
    #pragma unroll
    for (int i = 0; i < 8; ++i) {
        int row = tile_m * 16 + rbase + i;
        out[row * D_OUT + col] = elu1(acc[i] + bj);
    }
}

// ------------- graph pooling scatter: sums[batch[n]] += h[n]; cnt[batch[n]] += 1 -------------
__global__ void pool_scatter(const float* __restrict__ h, const int* __restrict__ batch,
                             float* __restrict__ gsum, float* __restrict__ gcnt) {
    int idx = blockIdx.x * blockDim.x + threadIdx.x;     // over N_NODES*64
    if (idx >= N_NODES * 64) return;
    int n = idx >> 6;
    int f = idx & 63;
    int g = batch[n];
    atomicAdd(&gsum[g * 64 + f], h[idx]);
    if (f == 0) atomicAdd(&gcnt[g], 1.0f);
}

// ------------- MLP head + log_softmax: one 64-thread block per graph -------------
__global__ void mlp_head(const float* __restrict__ gsum, const float* __restrict__ gcnt,
                         const float* __restrict__ fw1, const float* __restrict__ fb1,
                         const float* __restrict__ fw2, const float* __restrict__ fb2,
                         const float* __restrict__ fw3, const float* __restrict__ fb3,
                         float* __restrict__ out) {
    __shared__ float p[64], z1[64], z2[32], z3[10], lse;
    const int g = blockIdx.x;
    const int t = threadIdx.x;

    float cnt = fmaxf(gcnt[g], 1.0f);
    p[t] = gsum[g * 64 + t] / cnt;
    __syncthreads();

    float s = fb1[t];
    #pragma unroll 8
    for (int k = 0; k < 64; ++k) s += p[k] * fw1[k * 64 + t];
    z1[t] = elu1(s);
    __syncthreads();

    if (t < 32) {
        float s2 = fb2[t];
        #pragma unroll 8
        for (int k = 0; k < 64; ++k) s2 += z1[k] * fw2[k * 32 + t];
        z2[t] = elu1(s2);
    }
    __syncthreads();

    if (t < 10) {
        float s3 = fb3[t];
        #pragma unroll
        for (int k = 0; k < 32; ++k) s3 += z2[k] * fw3[k * 10 + t];
        z3[t] = s3;
    }
    __syncthreads();

    if (t == 0) {
        float mx = z3[0];
        for (int j = 1; j < 10; ++j) mx = fmaxf(mx, z3[j]);
        float se = 0.0f;
        for (int j = 0; j < 10; ++j) se += expf(z3[j] - mx);
        lse = mx + logf(se);
    }
    __syncthreads();

    if (t < 10) out[g * 10 + t] = z3[t] - lse;
}

// ============================= launcher =============================
extern "C" void kernel_launch(void* const* d_in, const int* in_sizes, int n_in,
                              void* d_out, int out_size, void* d_ws, size_t ws_size,
                              hipStream_t stream) {
    const float* x       = (const float*)d_in[0];
    const int*   ei      = (const int*)d_in[1];
    const int*   batch   = (const int*)d_in[2];
    const float* w1_rel  = (const float*)d_in[3];
    const float* b1      = (const float*)d_in[4];
    const float* w1_root = (const float*)d_in[5];
    const float* w2_rel  = (const float*)d_in[6];
    const float* b2      = (const float*)d_in[7];
    const float* w2_root = (const float*)d_in[8];
    const float* w3_rel  = (const float*)d_in[9];
    const float* b3      = (const float*)d_in[10];
    const float* w3_root = (const float*)d_in[11];
    const float* fw1     = (const float*)d_in[12];
    const float* fb1     = (const float*)d_in[13];
    const float* fw2     = (const float*)d_in[14];
    const float* fb2     = (const float*)d_in[15];
    const float* fw3     = (const float*)d_in[16];
    const float* fb3     = (const float*)d_in[17];
    float* out = (float*)d_out;

    const int* src = ei;
    const int* dst = ei + N_EDGES;

    // workspace layout (floats)
    float* ws   = (float*)d_ws;
    float* agg  = ws;                      // 100000*64
    float* h1   = agg  + N_NODES * 64;     // 100000*32
    float* h2   = h1   + N_NODES * 32;     // 100000*64
    float* h3   = h2   + N_NODES * 64;     // 100000*64
    float* gsum = h3   + N_NODES * 64;     // 512*64
    float* gcnt = gsum + N_GRAPHS * 64;    // 512

    const int TB = 256;
    auto blocks = [](int n, int tb) { return (n + tb - 1) / tb; };

    // ---- Layer 1: d_in=32 -> d_out=32 ----
    zero_f32<<<blocks(N_NODES * 32, TB), TB, 0, stream>>>(agg, N_NODES * 32);
    scatter_add<<<blocks(N_EDGES * 32, TB), TB, 0, stream>>>(x, agg, src, dst, N_EDGES, 5);
    {
        int waves = (N_NODES / 16) * (32 / 16);
        conv_wmma<32, 32><<<blocks(waves, 8), TB, 0, stream>>>(agg, x, w1_rel, w1_root, b1, h1, N_NODES);
    }

    // ---- Layer 2: 32 -> 64 ----
    zero_f32<<<blocks(N_NODES * 32, TB), TB, 0, stream>>>(agg, N_NODES * 32);
    scatter_add<<<blocks(N_EDGES * 32, TB), TB, 0, stream>>>(h1, agg, src, dst, N_EDGES, 5);
    {
        int waves = (N_NODES / 16) * (64 / 16);
        conv_wmma<32, 64><<<blocks(waves, 8), TB, 0, stream>>>(agg, h1, w2_rel, w2_root, b2, h2, N_NODES);
    }

    // ---- Layer 3: 64 -> 64 ----
    zero_f32<<<blocks(N_NODES * 64, TB), TB, 0, stream>>>(agg, N_NODES * 64);
    scatter_add<<<blocks(N_EDGES * 64, TB), TB, 0, stream>>>(h2, agg, src, dst, N_EDGES, 6);
    {
        int waves = (N_NODES / 16) * (64 / 16);
        conv_wmma<64, 64><<<blocks(waves, 8), TB, 0, stream>>>(agg, h2, w3_rel, w3_root, b3, h3, N_NODES);
    }

    // ---- mean pool over graphs ----
    zero_f32<<<blocks(N_GRAPHS * 64 + N_GRAPHS, TB), TB, 0, stream>>>(gsum, N_GRAPHS * 64 + N_GRAPHS);
    pool_scatter<<<blocks(N_NODES * 64, TB), TB, 0, stream>>>(h3, batch, gsum, gcnt);

    // ---- MLP head + log_softmax ----
    mlp_head<<<N_GRAPHS, 64, 0, stream>>>(gsum, gcnt, fw1, fb1, fw2, fb2, fw3, fb3, out);
}